// MACTitanLayer_64630667870915
// MI455X (gfx1250) — compile-verified
//
#include <hip/hip_runtime.h>
#include <hip/hip_bf16.h>
#include <math.h>

// ---------------------------------------------------------------------------
// MAC Titan layer for MI455X (gfx1250, wave32).
// Roofline: dominated by streaming final_w (736 MB fp32) -> ~32us @ 23.3TB/s.
// That matmul is a dedicated bandwidth GEMV (M=8). All other dense GEMMs run
// on V_WMMA_F32_16X16X4_F32 (full fp32 precision; shapes all tile 16x16xK4).
// ---------------------------------------------------------------------------

#define B_    8
#define S_    96
#define H_    96
#define PM_   16
#define FF_   2048
#define NH_   2
#define L_    208          // PM + 2S
#define ROWS_ (B_*S_)      // 768
#define LB_   (B_*L_)      // 1664
#define HD_   48
#define KIN_  (L_*H_)      // 19968
#define KOUT_ (S_*H_)      // 9216
#define NTOT_ (ROWS_*H_)   // 73728

typedef __attribute__((ext_vector_type(2))) float v2f;
typedef __attribute__((ext_vector_type(8))) float v8f;

__device__ __forceinline__ float wsum(float v) {
#pragma unroll
  for (int o = 16; o > 0; o >>= 1) v += __shfl_xor(v, o, 32);
  return v;
}
__device__ __forceinline__ float wmaxr(float v) {
#pragma unroll
  for (int o = 16; o > 0; o >>= 1) v = fmaxf(v, __shfl_xor(v, o, 32));
  return v;
}
__device__ __forceinline__ float sigm(float x) { return 1.f / (1.f + __expf(-x)); }
__device__ __forceinline__ float siluf(float x) { return x * sigm(x); }

// ---------------------------------------------------------------------------
// Generic batched WMMA GEMM:  C = act(A? @ B? + bias)
//   A logical [M,K] (TRA: physical [K,M], lda = phys row stride)
//   B logical [K,N] (TRB: physical [N,K], ldb = phys row stride)
// One 16x16 C tile per wave; K stepped by 4 through V_WMMA_F32_16X16X4_F32.
// Requires M%16==0, N%16==0, K%4==0 (true for every call site here).
// Batch: z = blockIdx.z; offset = (z/bdiv)*sXo + (z%bdiv)*sXi.
// ---------------------------------------------------------------------------
template <bool TRA, bool TRB, int ACT>
__global__ __launch_bounds__(128) void wmma_gemm_f32(
    const float* __restrict__ A, const float* __restrict__ B,
    const float* __restrict__ bias, float* __restrict__ C,
    int M, int N, int K, int lda, int ldb, int ldc,
    int bdiv, int sAo, int sAi, int sBo, int sBi, int sCo, int sCi) {
  const int z = blockIdx.z;
  A += (z / bdiv) * sAo + (z % bdiv) * sAi;
  B += (z / bdiv) * sBo + (z % bdiv) * sBi;
  C += (z / bdiv) * sCo + (z % bdiv) * sCi;

  const int nt = N >> 4;
  const int tiles = (M >> 4) * nt;
  const int tile = blockIdx.x * 4 + (threadIdx.x >> 5);
  if (tile >= tiles) return;                 // wave-uniform (EXEC all-ones inside)
  const int m0 = (tile / nt) << 4;
  const int n0 = (tile % nt) << 4;
  const int lane = threadIdx.x & 31;
  const int half = lane >> 4;                // K-half select (ISA 16x4 f32 A layout)
  const int r16 = lane & 15;                 // M row (A) / N col (B,C)

  v8f acc = {};
  for (int k0 = 0; k0 < K; k0 += 4) {
    v2f a, b;
    const int ka = k0 + 2 * half;            // lanes 0-15: K=k0,k0+1; 16-31: K=k0+2,k0+3
    if (!TRA) {
      a.x = A[(m0 + r16) * lda + ka];
      a.y = A[(m0 + r16) * lda + ka + 1];
    } else {
      a.x = A[ka * lda + (m0 + r16)];
      a.y = A[(ka + 1) * lda + (m0 + r16)];
    }
    const int kb = k0 + half;                // B: VGPR v holds K row (k0+2v+half), lane = N
    if (!TRB) {
      b.x = B[kb * ldb + (n0 + r16)];
      b.y = B[(kb + 2) * ldb + (n0 + r16)];
    } else {
      b.x = B[(n0 + r16) * ldb + kb];
      b.y = B[(n0 + r16) * ldb + kb + 2];
    }
    acc = __builtin_amdgcn_wmma_f32_16x16x4_f32(false, a, false, b, (short)0, acc,
                                                false, false);
  }
#pragma unroll
  for (int r = 0; r < 8; ++r) {              // C/D: VGPR r -> M=r (+8 for upper half)
    const int m = m0 + half * 8 + r;
    const int n = n0 + r16;
    float v = acc[r];
    if (bias) v += bias[n];
    if (ACT == 1) v = siluf(v);
    C[m * ldc + n] = v;
  }
}

template <bool TRA, bool TRB, int ACT>
static void launch_gemm(const float* A, const float* B, const float* bias, float* C,
                        int M, int N, int K, int lda, int ldb, int ldc,
                        hipStream_t s, int nz = 1, int bdiv = 1,
                        int sAo = 0, int sAi = 0, int sBo = 0, int sBi = 0,
                        int sCo = 0, int sCi = 0) {
  const int tiles = (M / 16) * (N / 16);
  dim3 grid((tiles + 3) / 4, 1, nz);
  wmma_gemm_f32<TRA, TRB, ACT><<<grid, 128, 0, s>>>(A, B, bias, C, M, N, K, lda, ldb,
                                                    ldc, bdiv, sAo, sAi, sBo, sBi, sCo, sCi);
}

// ---------------------------------------------------------------------------
// Bandwidth streamer for xf = xe[8,19968] @ final_w.T + final_b.
// One wave per output row j: coalesced float4 reads of final_w row (78 KB,
// read exactly once -> 736 MB total HBM), xe rows stay hot in L1/L2.
// ---------------------------------------------------------------------------
__global__ __launch_bounds__(256) void k_final_gemv(const float* __restrict__ xe,
                                                    const float* __restrict__ W,
                                                    const float* __restrict__ bias,
                                                    float* __restrict__ xf) {
  const int wave = threadIdx.x >> 5, lane = threadIdx.x & 31;
  const int j = blockIdx.x * 8 + wave;
  if (j >= KOUT_) return;
  const float* wr = W + (size_t)j * KIN_;
  float acc[B_] = {};
  for (int k = lane * 4; k < KIN_; k += 128) {
    const float4 wv = *(const float4*)(wr + k);
#pragma unroll
    for (int b = 0; b < B_; ++b) {
      const float4 xv = *(const float4*)(xe + b * KIN_ + k);
      acc[b] += wv.x * xv.x + wv.y * xv.y + wv.z * xv.z + wv.w * xv.w;
    }
  }
#pragma unroll
  for (int b = 0; b < B_; ++b) {
    const float r = wsum(acc[b]);
    if (lane == 0) xf[b * KOUT_ + j] = r + bias[j];
  }
}

// ---- row-wise L2 normalize (+ optional SiLU); width 96, one wave per row ---
__global__ __launch_bounds__(128) void k_l2norm(const float* __restrict__ in,
                                                float* __restrict__ out,
                                                int rows, int dosilu) {
  const int row = blockIdx.x * 4 + (threadIdx.x >> 5);
  if (row >= rows) return;
  const int lane = threadIdx.x & 31;
  const float* p = in + row * H_;
  float t0 = p[lane], t1 = p[lane + 32], t2 = p[lane + 64];
  float ss = wsum(t0 * t0 + t1 * t1 + t2 * t2);
  const float inv = 1.f / fmaxf(sqrtf(ss), 1e-12f);
  float o0 = t0 * inv, o1 = t1 * inv, o2 = t2 * inv;
  if (dosilu) { o0 = siluf(o0); o1 = siluf(o1); o2 = siluf(o2); }
  float* q = out + row * H_;
  q[lane] = o0; q[lane + 32] = o1; q[lane + 64] = o2;
}

// ---- residual add + LayerNorm (+ optional SiLU); width 96 -----------------
__global__ __launch_bounds__(128) void k_add_ln(const float* __restrict__ a,
                                                const float* __restrict__ r,
                                                const float* __restrict__ w,
                                                const float* __restrict__ b,
                                                float* __restrict__ out,
                                                int rows, int dosilu) {
  const int row = blockIdx.x * 4 + (threadIdx.x >> 5);
  if (row >= rows) return;
  const int lane = threadIdx.x & 31;
  const float* ap = a + row * H_;
  const float* rp = r + row * H_;
  float t[3];
#pragma unroll
  for (int i = 0; i < 3; ++i) t[i] = ap[lane + 32 * i] + rp[lane + 32 * i];
  const float mean = wsum(t[0] + t[1] + t[2]) * (1.f / H_);
  float vs = 0.f;
#pragma unroll
  for (int i = 0; i < 3; ++i) { const float d = t[i] - mean; vs += d * d; }
  const float inv = rsqrtf(wsum(vs) * (1.f / H_) + 1e-5f);
  float* q = out + row * H_;
#pragma unroll
  for (int i = 0; i < 3; ++i) {
    const int c = lane + 32 * i;
    float o = (t[i] - mean) * inv * w[c] + b[c];
    if (dosilu) o = siluf(o);
    q[c] = o;
  }
}

// ---- in-place row softmax with pre-scale; one wave per row ----------------
__global__ __launch_bounds__(128) void k_softmax(float* __restrict__ sc, int rows,
                                                 int len, float scale) {
  const int row = blockIdx.x * 4 + (threadIdx.x >> 5);
  if (row >= rows) return;
  const int lane = threadIdx.x & 31;
  float* p = sc + (size_t)row * len;
  float m = -1e30f;
  for (int c = lane; c < len; c += 32) m = fmaxf(m, p[c] * scale);
  m = wmaxr(m);
  float s = 0.f;
  for (int c = lane; c < len; c += 32) { const float e = __expf(p[c] * scale - m); p[c] = e; s += e; }
  s = wsum(s);
  const float inv = 1.f / s;
  for (int c = lane; c < len; c += 32) p[c] *= inv;
}

// ---- build xc = concat(pm, nmm, x) along sequence ------------------------
__global__ __launch_bounds__(256) void k_concat(const float* __restrict__ pm,
                                                const float* __restrict__ nmm,
                                                const float* __restrict__ x,
                                                float* __restrict__ xc) {
  const int i = blockIdx.x * 256 + threadIdx.x;
  if (i >= LB_ * H_) return;
  const int c = i % H_;
  const int l = (i / H_) % L_;
  const int b = i / (H_ * L_);
  float v;
  if (l < PM_)            v = pm[l * H_ + c];
  else if (l < PM_ + S_)  v = nmm[(b * S_ + (l - PM_)) * H_ + c];
  else                    v = x[(b * S_ + (l - PM_ - S_)) * H_ + c];
  xc[i] = v;
}

// ---- elementwise helpers ---------------------------------------------------
__global__ void k_silu(const float* __restrict__ z, float* __restrict__ h, int n) {
  const int i = blockIdx.x * 256 + threadIdx.x;
  if (i < n) h[i] = siluf(z[i]);
}
__global__ void k_scalediff(const float* __restrict__ p, const float* __restrict__ v,
                            float* __restrict__ d, int n, float s) {
  const int i = blockIdx.x * 256 + threadIdx.x;
  if (i < n) d[i] = (p[i] - v[i]) * s;
}
__global__ void k_dsilu_mul(const float* __restrict__ dh, const float* __restrict__ z,
                            float* __restrict__ dz, int n) {
  const int i = blockIdx.x * 256 + threadIdx.x;
  if (i < n) {
    const float zz = z[i], s = sigm(zz);
    dz[i] = dh[i] * (s * (1.f + zz * (1.f - s)));
  }
}
__global__ void k_update(const float* __restrict__ p, const float* __restrict__ g,
                         float* __restrict__ np, int n) {
  const int i = blockIdx.x * 256 + threadIdx.x;
  if (i < n) np[i] = 0.999f * p[i] - 0.3f * g[i];
}
__global__ void k_gate(const float* __restrict__ xf, const float* __restrict__ y,
                       float* __restrict__ out, int n) {
  const int i = blockIdx.x * 256 + threadIdx.x;
  if (i < n) out[i] = xf[i] * sigm(y[i]);
}
__global__ __launch_bounds__(32) void k_colsum(const float* __restrict__ D,
                                               float* __restrict__ out,
                                               int rows, int cols) {
  const int c = blockIdx.x, lane = threadIdx.x;
  float s = 0.f;
  for (int r = lane; r < rows; r += 32) s += D[(size_t)r * cols + c];
  s = wsum(s);
  if (lane == 0) out[c] = s;
}

// ---------------------------------------------------------------------------
extern "C" void kernel_launch(void* const* d_in, const int* in_sizes, int n_in,
                              void* d_out, int out_size, void* d_ws, size_t ws_size,
                              hipStream_t stream) {
  (void)in_sizes; (void)n_in; (void)out_size; (void)ws_size;
  const float* X    = (const float*)d_in[0];   // [8,96,96]
  const float* PMm  = (const float*)d_in[1];   // [16,96]
  const float* q_w  = (const float*)d_in[2];   const float* q_b  = (const float*)d_in[3];
  const float* ipw  = (const float*)d_in[4];   const float* ipb  = (const float*)d_in[5];
  const float* opw  = (const float*)d_in[6];   const float* opb  = (const float*)d_in[7];
  const float* l1w  = (const float*)d_in[8];   const float* l1b  = (const float*)d_in[9];
  const float* l2w  = (const float*)d_in[10];  const float* l2b  = (const float*)d_in[11];
  const float* f1w  = (const float*)d_in[12];  const float* f1b  = (const float*)d_in[13];
  const float* f2w  = (const float*)d_in[14];  const float* f2b  = (const float*)d_in[15];
  const float* fw   = (const float*)d_in[16];  const float* fb   = (const float*)d_in[17];
  const float* k_w  = (const float*)d_in[18];  const float* k_b  = (const float*)d_in[19];
  const float* v_w  = (const float*)d_in[20];  const float* v_b  = (const float*)d_in[21];
  const float* m1w  = (const float*)d_in[22];  const float* m1b  = (const float*)d_in[23];
  const float* m2w  = (const float*)d_in[24];  const float* m2b  = (const float*)d_in[25];
  float* out = (float*)d_out;

  // workspace bump allocator (~28 MB fp32 total)
  float* ws = (float*)d_ws;
  size_t off = 0;
  auto alloc = [&](size_t n) { float* p = ws + off; off += n; return p; };
  float* q0    = alloc(NTOT_);
  float* qrs   = alloc(NTOT_);
  float* h1    = alloc(ROWS_ * 2 * H_);
  float* nmm   = alloc(NTOT_);
  float* xc    = alloc(LB_ * H_);
  float* qkv   = alloc(LB_ * 3 * H_);
  float* sc    = alloc(B_ * NH_ * L_ * L_);
  float* attno = alloc(LB_ * H_);
  float* o2    = alloc(LB_ * H_);
  float* x1    = alloc(LB_ * H_);
  float* ffh   = alloc((size_t)LB_ * FF_);
  float* ff2   = alloc(LB_ * H_);
  float* xe    = alloc(LB_ * H_);
  float* xf    = alloc(NTOT_);
  float* kp    = alloc(NTOT_);
  float* vp    = alloc(NTOT_);
  float* z1    = alloc(ROWS_ * 2 * H_);
  float* hh    = alloc(ROWS_ * 2 * H_);
  float* pred  = alloc(NTOT_);
  float* dpred = alloc(NTOT_);
  float* dh    = alloc(ROWS_ * 2 * H_);
  float* dz1   = alloc(ROWS_ * 2 * H_);
  float* g2w   = alloc(H_ * 2 * H_);
  float* g2b   = alloc(H_);
  float* g1w   = alloc(2 * H_ * H_);
  float* g1b   = alloc(2 * H_);
  float* n1w   = alloc(2 * H_ * H_);
  float* n1b   = alloc(2 * H_);
  float* n2w   = alloc(H_ * 2 * H_);
  float* n2b   = alloc(H_);
  float* q2    = alloc(NTOT_);
  float* qn    = alloc(NTOT_);
  float* h2    = alloc(ROWS_ * 2 * H_);
  float* yb    = alloc(NTOT_);

  // ---- neural-memory retrieval on x ----
  launch_gemm<false, true, 0>(X, q_w, q_b, q0, ROWS_, H_, H_, H_, H_, H_, stream);
  k_l2norm<<<ROWS_ / 4, 128, 0, stream>>>(q0, qrs, ROWS_, 1);
  launch_gemm<false, true, 1>(qrs, m1w, m1b, h1, ROWS_, 2 * H_, H_, H_, H_, 2 * H_, stream);
  launch_gemm<false, true, 0>(h1, m2w, m2b, nmm, ROWS_, H_, 2 * H_, 2 * H_, 2 * H_, H_, stream);

  // ---- concat [pm, nmm, x] -> xc [8,208,96] ----
  k_concat<<<(LB_ * H_ + 255) / 256, 256, 0, stream>>>(PMm, nmm, X, xc);

  // ---- transformer encoder layer ----
  launch_gemm<false, true, 0>(xc, ipw, ipb, qkv, LB_, 3 * H_, H_, H_, H_, 3 * H_, stream);
  // scores = Q @ K^T, batched over 16 (b,h)
  launch_gemm<false, true, 0>(qkv, qkv + H_, nullptr, sc, L_, L_, HD_, 3 * H_, 3 * H_, L_,
                              stream, B_ * NH_, NH_, L_ * 3 * H_, HD_, L_ * 3 * H_, HD_,
                              NH_ * L_ * L_, L_ * L_);
  k_softmax<<<(B_ * NH_ * L_ + 3) / 4, 128, 0, stream>>>(sc, B_ * NH_ * L_, L_,
                                                         0.14433756729740643f); // 1/sqrt(48)
  // attn out = P @ V
  launch_gemm<false, false, 0>(sc, qkv + 2 * H_, nullptr, attno, L_, HD_, L_, L_, 3 * H_, H_,
                               stream, B_ * NH_, NH_, NH_ * L_ * L_, L_ * L_,
                               L_ * 3 * H_, HD_, L_ * H_, HD_);
  launch_gemm<false, true, 0>(attno, opw, opb, o2, LB_, H_, H_, H_, H_, H_, stream);
  k_add_ln<<<LB_ / 4, 128, 0, stream>>>(xc, o2, l1w, l1b, x1, LB_, 0);
  launch_gemm<false, true, 1>(x1, f1w, f1b, ffh, LB_, FF_, H_, H_, H_, FF_, stream);
  launch_gemm<false, true, 0>(ffh, f2w, f2b, ff2, LB_, H_, FF_, FF_, FF_, H_, stream);
  k_add_ln<<<LB_ / 4, 128, 0, stream>>>(x1, ff2, l2w, l2b, xe, LB_, 1); // fused SiLU -> xe

  // ---- the 736 MB streamer: xf = xe @ final_w^T + final_b ----
  k_final_gemv<<<KOUT_ / 8, 256, 0, stream>>>(xe, fw, fb, xf);

  // ---- test-time-training step on memory MLP ----
  launch_gemm<false, true, 0>(xf, k_w, k_b, kp, ROWS_, H_, H_, H_, H_, H_, stream);
  launch_gemm<false, true, 0>(xf, v_w, v_b, vp, ROWS_, H_, H_, H_, H_, H_, stream);
  launch_gemm<false, true, 0>(kp, m1w, m1b, z1, ROWS_, 2 * H_, H_, H_, H_, 2 * H_, stream);
  k_silu<<<(ROWS_ * 2 * H_ + 255) / 256, 256, 0, stream>>>(z1, hh, ROWS_ * 2 * H_);
  launch_gemm<false, true, 0>(hh, m2w, m2b, pred, ROWS_, H_, 2 * H_, 2 * H_, 2 * H_, H_, stream);
  k_scalediff<<<(NTOT_ + 255) / 256, 256, 0, stream>>>(pred, vp, dpred, NTOT_, 2.f / NTOT_);
  launch_gemm<true, false, 0>(dpred, hh, nullptr, g2w, H_, 2 * H_, ROWS_, H_, 2 * H_, 2 * H_, stream);
  k_colsum<<<H_, 32, 0, stream>>>(dpred, g2b, ROWS_, H_);
  launch_gemm<false, false, 0>(dpred, m2w, nullptr, dh, ROWS_, 2 * H_, H_, H_, 2 * H_, 2 * H_, stream);
  k_dsilu_mul<<<(ROWS_ * 2 * H_ + 255) / 256, 256, 0, stream>>>(dh, z1, dz1, ROWS_ * 2 * H_);
  launch_gemm<true, false, 0>(dz1, kp, nullptr, g1w, 2 * H_, H_, ROWS_, 2 * H_, H_, H_, stream);
  k_colsum<<<2 * H_, 32, 0, stream>>>(dz1, g1b, ROWS_, 2 * H_);
  k_update<<<(2 * H_ * H_ + 255) / 256, 256, 0, stream>>>(m1w, g1w, n1w, 2 * H_ * H_);
  k_update<<<1, 256, 0, stream>>>(m1b, g1b, n1b, 2 * H_);
  k_update<<<(2 * H_ * H_ + 255) / 256, 256, 0, stream>>>(m2w, g2w, n2w, 2 * H_ * H_);
  k_update<<<1, 256, 0, stream>>>(m2b, g2b, n2b, H_);

  // ---- retrieval with updated weights + sigmoid gate ----
  launch_gemm<false, true, 0>(xf, q_w, q_b, q2, ROWS_, H_, H_, H_, H_, H_, stream);
  k_l2norm<<<ROWS_ / 4, 128, 0, stream>>>(q2, qn, ROWS_, 0);
  launch_gemm<false, true, 1>(qn, n1w, n1b, h2, ROWS_, 2 * H_, H_, H_, H_, 2 * H_, stream);
  launch_gemm<false, true, 0>(h2, n2w, n2b, yb, ROWS_, H_, 2 * H_, 2 * H_, 2 * H_, H_, stream);
  k_gate<<<(NTOT_ + 255) / 256, 256, 0, stream>>>(xf, yb, out, NTOT_);
}